// _GATLayer_18605798326408
// MI455X (gfx1250) — compile-verified
//
#include <hip/hip_runtime.h>
#include <cstdint>

#define B_   8
#define N_   2048
#define F_   64
#define ALPHA 0.2f

typedef float v2f __attribute__((ext_vector_type(2)));
typedef float v8f __attribute__((ext_vector_type(8)));

#define LDS_STRIDE 72   // 2*72 % 64 == 16 -> half-wave row reads hit disjoint bank groups

__device__ __forceinline__ float lrelu(float x) { return x > 0.f ? x : ALPHA * x; }

// ---------------------------------------------------------------------------
// Kernel 1: Wh[b,n,:] = h[b,n,:] @ W   via v_wmma_f32_16x16x4_f32
// block = 128 threads (4 waves), each wave computes a 16x64 tile of Wh.
// grid  = B * (N/64)
// ---------------------------------------------------------------------------
__global__ __launch_bounds__(128) void k_wh(const float* __restrict__ h,
                                            const float* __restrict__ W,
                                            float* __restrict__ Wh) {
  __shared__ float Wl[64 * LDS_STRIDE];
  const int tid = threadIdx.x;
  for (int i = tid; i < 64 * 64; i += 128) {
    int r = i >> 6, c = i & 63;
    Wl[r * LDS_STRIDE + c] = W[i];
  }
  __syncthreads();

  const int b      = blockIdx.x >> 5;   // N/64 = 32 row-blocks per batch
  const int rowblk = blockIdx.x & 31;
  const int wave   = tid >> 5;
  const int lane   = tid & 31;
  const int row0   = rowblk * 64 + wave * 16;
  const int m      = lane & 15;
  const int hi     = lane >> 4;
  const int kk     = hi * 2;

  const float* hrow = h + ((size_t)b * N_ + row0 + m) * F_;

  v8f acc[4] = {};  // output column tiles n0 = 0,16,32,48

  for (int k0 = 0; k0 < F_; k0 += 4) {
    v2f a;
    a.x = hrow[k0 + kk];
    a.y = hrow[k0 + kk + 1];
#pragma unroll
    for (int t = 0; t < 4; ++t) {
      v2f bv;
      bv.x = Wl[(k0 + kk) * LDS_STRIDE + t * 16 + m];
      bv.y = Wl[(k0 + kk + 1) * LDS_STRIDE + t * 16 + m];
      acc[t] = __builtin_amdgcn_wmma_f32_16x16x4_f32(false, a, false, bv,
                                                     (short)0, acc[t], false, false);
    }
  }

#pragma unroll
  for (int v = 0; v < 8; ++v) {
    const size_t orow = ((size_t)b * N_ + row0 + v + hi * 8) * F_;
#pragma unroll
    for (int t = 0; t < 4; ++t)
      Wh[orow + t * 16 + m] = acc[t][v];
  }
}

// ---------------------------------------------------------------------------
// Kernel 2: s_src[r] = Wh[r,:]·a_src ; s_dst[r] = Wh[r,:]·a_dst
// ---------------------------------------------------------------------------
__global__ __launch_bounds__(256) void k_scores(const float* __restrict__ Wh,
                                                const float* __restrict__ a,
                                                float* __restrict__ s_src,
                                                float* __restrict__ s_dst) {
  __shared__ float al[2 * F_];
  const int tid = threadIdx.x;
  if (tid < 2 * F_) al[tid] = a[tid];
  __syncthreads();

  const int row = blockIdx.x * 256 + tid;   // 0 .. B*N-1
  const float* w = Wh + (size_t)row * F_;
  float ss = 0.f, sd = 0.f;
#pragma unroll 8
  for (int o = 0; o < F_; ++o) {
    const float v = w[o];
    ss += v * al[o];
    sd += v * al[F_ + o];
  }
  s_src[row] = ss;
  s_dst[row] = sd;
}

// ---------------------------------------------------------------------------
// Kernel 3: maxd[b] = max_j s_dst[b,j]   (one block per batch)
// ---------------------------------------------------------------------------
__global__ __launch_bounds__(256) void k_max(const float* __restrict__ s_dst,
                                             float* __restrict__ maxd) {
  __shared__ float red[256];
  const int b = blockIdx.x, tid = threadIdx.x;
  float mx = -3.402823466e38f;
  for (int j = tid; j < N_; j += 256) mx = fmaxf(mx, s_dst[(size_t)b * N_ + j]);
  red[tid] = mx;
  __syncthreads();
  for (int s = 128; s > 0; s >>= 1) {
    if (tid < s) red[tid] = fmaxf(red[tid], red[tid + s]);
    __syncthreads();
  }
  if (tid == 0) maxd[b] = red[0];
}

// ---------------------------------------------------------------------------
// Kernel 4: flash-style  out = elu( softmax(lrelu(s_src_i + s_dst_j)) @ Wh )
// block = 128 threads (4 waves); each wave owns a 16-row i-tile; the block
// streams 64-row j-chunks of Wh through LDS shared by all 4 waves.
// grid  = B * (N/64)
// ---------------------------------------------------------------------------
__global__ __launch_bounds__(128) void k_attn(const float* __restrict__ Wh,
                                              const float* __restrict__ s_src,
                                              const float* __restrict__ s_dst,
                                              const float* __restrict__ maxd,
                                              float* __restrict__ out) {
  __shared__ float Wl[64 * LDS_STRIDE];
  __shared__ float sdl[64];

  const int tid    = threadIdx.x;
  const int b      = blockIdx.x >> 5;
  const int rowblk = blockIdx.x & 31;
  const int wave   = tid >> 5;
  const int lane   = tid & 31;
  const int row0   = rowblk * 64 + wave * 16;
  const int m      = lane & 15;
  const int hi     = lane >> 4;
  const int kk     = hi * 2;

  const float ssrc = s_src[(size_t)b * N_ + row0 + m];
  const float mi   = lrelu(ssrc + maxd[b]);   // exact row max (lrelu is monotone)

  v8f acc[4] = {};
  float z = 0.f;

  for (int j0 = 0; j0 < N_; j0 += 64) {
    __syncthreads();
    // Stage 64x64 Wh chunk (float4) + 64 s_dst values into LDS.
    for (int i = tid; i < 64 * 16; i += 128) {
      const int r = i >> 4, c4 = (i & 15) << 2;
      const float4 v = *(const float4*)(Wh + ((size_t)b * N_ + j0 + r) * F_ + c4);
      *(float4*)(&Wl[r * LDS_STRIDE + c4]) = v;
    }
    if (tid < 64) sdl[tid] = s_dst[(size_t)b * N_ + j0 + tid];
    if (j0 + 64 < N_)  // hint next chunk into cache (global_prefetch_b8)
      __builtin_prefetch(Wh + ((size_t)b * N_ + j0 + 64 + (tid >> 1)) * F_ + (tid & 1) * 32, 0, 0);
    __syncthreads();

#pragma unroll 4
    for (int jj = 0; jj < 64; jj += 4) {
      // probability fragment P[16 x 4] for this wave's i-tile
      const float sd0 = sdl[jj + kk];
      const float sd1 = sdl[jj + kk + 1];
      const float p0 = __expf(lrelu(ssrc + sd0) - mi);
      const float p1 = __expf(lrelu(ssrc + sd1) - mi);
      z += p0 + p1;
      v2f a; a.x = p0; a.y = p1;
#pragma unroll
      for (int t = 0; t < 4; ++t) {
        v2f bv;
        bv.x = Wl[(jj + kk) * LDS_STRIDE + t * 16 + m];
        bv.y = Wl[(jj + kk + 1) * LDS_STRIDE + t * 16 + m];
        acc[t] = __builtin_amdgcn_wmma_f32_16x16x4_f32(false, a, false, bv,
                                                       (short)0, acc[t], false, false);
      }
    }
  }

  // Z[row m] = z(lane m) + z(lane m^16): full softmax denominator per row.
  const float z2 = z + __shfl_xor(z, 16, 32);

#pragma unroll
  for (int v = 0; v < 8; ++v) {
    const float Z = __shfl(z2, v + hi * 8, 32);   // lane s<16 holds Z[row0+s]
    const float rz = 1.f / Z;
    const size_t orow = ((size_t)b * N_ + row0 + v + hi * 8) * F_;
#pragma unroll
    for (int t = 0; t < 4; ++t) {
      float x = acc[t][v] * rz;
      x = x > 0.f ? x : __expf(x) - 1.f;   // ELU
      out[orow + t * 16 + m] = x;
    }
  }
}

// ---------------------------------------------------------------------------
extern "C" void kernel_launch(void* const* d_in, const int* in_sizes, int n_in,
                              void* d_out, int out_size, void* d_ws, size_t ws_size,
                              hipStream_t stream) {
  const float* h = (const float*)d_in[0];   // (B, N, F)
  const float* W = (const float*)d_in[1];   // (F, F)
  const float* a = (const float*)d_in[2];   // (2F, 1)
  float* out = (float*)d_out;               // (B, N, F)

  float* Wh    = (float*)d_ws;                       // B*N*F floats (4 MB)
  float* s_src = Wh + (size_t)B_ * N_ * F_;          // B*N
  float* s_dst = s_src + (size_t)B_ * N_;            // B*N
  float* maxd  = s_dst + (size_t)B_ * N_;            // B

  k_wh    <<<B_ * (N_ / 64), 128, 0, stream>>>(h, W, Wh);
  k_scores<<<(B_ * N_) / 256, 256, 0, stream>>>(Wh, a, s_src, s_dst);
  k_max   <<<B_,             256, 0, stream>>>(s_dst, maxd);
  k_attn  <<<B_ * (N_ / 64), 128, 0, stream>>>(Wh, s_src, s_dst, maxd, out);
}